// SS2DBlock_91061896610110
// MI455X (gfx1250) — compile-verified
//
#include <hip/hip_runtime.h>
#include <math.h>

#define NSEQ   32
#define LSEQ   4096
#define NTOK   (NSEQ * LSEQ)   // 131072 tokens
#define CCH    64
#define DIN    128
#define DSTATE 16
#define DTRANK 4

typedef __attribute__((ext_vector_type(2))) float    v2f;
typedef __attribute__((ext_vector_type(4))) float    v4f;
typedef __attribute__((ext_vector_type(8))) float    v8f;
typedef __attribute__((ext_vector_type(4))) unsigned v4u;
typedef __attribute__((ext_vector_type(8))) unsigned v8u;

// CDNA5 fp32 WMMA: D(16x16,f32) = A(16x4,f32) x B(4x16,f32) + C
__device__ __forceinline__ v8f wmma4(v2f a, v2f b, v8f c) {
  return __builtin_amdgcn_wmma_f32_16x16x4_f32(false, a, false, b, (short)0, c,
                                               false, false);
}

__device__ __forceinline__ float siluf(float x) { return x / (1.0f + __expf(-x)); }

// ---------------------------------------------------------------------------
// gfx1250 async global->LDS copy (16B per lane), tracked by ASYNCcnt.
// GVS mode: mem = SADDR(base) + VADDR(byte offset); VDST = LDS byte address.
// ---------------------------------------------------------------------------
__device__ __forceinline__ void async_copy_b128(unsigned lds_addr, unsigned goff,
                                                const void* base) {
  asm volatile("global_load_async_to_lds_b128 %0, %1, %2"
               :: "v"(lds_addr), "v"(goff), "s"(base) : "memory");
}
__device__ __forceinline__ void wait_async() {
  asm volatile("s_wait_asynccnt 0x0" ::: "memory");
}

// ---------------------------------------------------------------------------
// gfx1250 Tensor Data Mover: DMA a contiguous fp32 block (<=65535 elems) from
// global memory into LDS. D# built per ISA ch.8 (group0 + group1, 2D tile with
// one row); issued by one wave, tracked by TENSORcnt.
// ---------------------------------------------------------------------------
__device__ __forceinline__ void tdm_load_contig(void* lds_dst, const void* gsrc,
                                                unsigned nelem) {
  unsigned lds = (unsigned)(unsigned long long)lds_dst;
  unsigned long long ga = (unsigned long long)gsrc;
  v4u g0;
  g0[0] = 1u;                                              // count=1 (valid user D#)
  g0[1] = lds;                                             // lds_addr
  g0[2] = (unsigned)ga;                                    // global_addr[31:0]
  g0[3] = ((unsigned)(ga >> 32) & 0x01FFFFFFu) | (2u << 30); // addr[56:32] | type=2
  v8u g1;
  g1[0] = 2u << 16;                                        // data_size = 4B
  g1[1] = (nelem & 0xFFFFu) << 16;                         // tensor_dim0[15:0]
  g1[2] = (nelem >> 16) | (1u << 16);                      // tensor_dim0[31:16] | tensor_dim1=1
  g1[3] = (nelem & 0xFFFFu) << 16;                         // tile_dim0
  g1[4] = 1u;                                              // tile_dim1=1, tile_dim2=0
  g1[5] = nelem;                                           // tensor_dim0_stride lo
  g1[6] = 0u;
  g1[7] = 0u;
  asm volatile("tensor_load_to_lds %0, %1" :: "s"(g0), "s"(g1) : "memory");
}

// ---------------------------------------------------------------------------
// K1: build 4 direction sequences + LayerNorm over C=64.
// ---------------------------------------------------------------------------
__global__ void k_prep_ln(const float* __restrict__ x, const float* __restrict__ ln_w,
                          const float* __restrict__ ln_b, float* __restrict__ normed) {
  int tid  = threadIdx.x;
  int lane = tid & 31;
  int T    = blockIdx.x * 8 + (tid >> 5);
  int s = T >> 12;
  int l = T & 4095;
  int b = s & 7;
  int dir = s >> 3;
  int lr = (dir >= 2) ? (4095 - l) : l;
  int h, w;
  if ((dir & 1) == 0) { h = lr >> 6; w = lr & 63; }
  else               { w = lr >> 6; h = lr & 63; }
  const float* xb = x + (size_t)b * (64 * 4096) + (size_t)h * 64 + w;
  float v0 = xb[(size_t)lane * 4096];
  float v1 = xb[(size_t)(lane + 32) * 4096];
  float sum = v0 + v1;
  float sq  = v0 * v0 + v1 * v1;
#pragma unroll
  for (int m = 1; m < 32; m <<= 1) {
    sum += __shfl_xor(sum, m, 32);
    sq  += __shfl_xor(sq,  m, 32);
  }
  float mu  = sum * (1.0f / 64.0f);
  float var = sq * (1.0f / 64.0f) - mu * mu;
  float rs  = rsqrtf(var + 1e-5f);
  float* dst = normed + (size_t)T * 64;
  dst[lane]      = (v0 - mu) * rs * ln_w[lane]      + ln_b[lane];
  dst[lane + 32] = (v1 - mu) * rs * ln_w[lane + 32] + ln_b[lane + 32];
}

// ---------------------------------------------------------------------------
// K2: in-projection GEMM. M=NTOK, N=256, K=64 via fp32 WMMA 16x16x4.
// W_in (64 KB) staged by TDM, A tile (16 KB) by async b128 loads.
// ---------------------------------------------------------------------------
__global__ void k_inproj(const float* __restrict__ normed, const float* __restrict__ W_in,
                         float* __restrict__ xs_pre, float* __restrict__ zbuf) {
  extern __shared__ float smem[];
  float* sW = smem;              // 256*64 floats (64 KB)
  float* sA = smem + 256 * 64;   // 64*64 floats (16 KB)
  int tid = threadIdx.x;
  int t0  = blockIdx.x * 64;

  if (tid < 32) tdm_load_contig(sW, W_in, 256 * 64);
  {
    unsigned lA = (unsigned)(unsigned long long)(void*)sA;
    unsigned g0 = (unsigned)blockIdx.x * (64u * 64u * 4u);
#pragma unroll
    for (unsigned i = 0; i < 4; ++i) {
      unsigned idx = (unsigned)tid + i * 256u;
      async_copy_b128(lA + idx * 16u, g0 + idx * 16u, normed);
    }
  }
  wait_async();
  if (tid < 32) __builtin_amdgcn_s_wait_tensorcnt(0);
  __syncthreads();

  int lane = tid & 31, wave = tid >> 5;
  int wm = wave & 3, wn = wave >> 2;
  int row = lane & 15, half = lane >> 4;
  const v8f vz = {0.f, 0.f, 0.f, 0.f, 0.f, 0.f, 0.f, 0.f};
  v8f acc[8];
#pragma unroll
  for (int j = 0; j < 8; ++j) acc[j] = vz;

  for (int k0 = 0; k0 < 64; k0 += 4) {
    int kk = k0 + 2 * half;
    v2f a = *(const v2f*)&sA[(wm * 16 + row) * 64 + kk];
#pragma unroll
    for (int j = 0; j < 8; ++j) {
      v2f bf = *(const v2f*)&sW[(wn * 128 + j * 16 + row) * 64 + kk];
      acc[j] = wmma4(a, bf, acc[j]);
    }
  }
  float* dst = (wn == 0) ? xs_pre : zbuf;
#pragma unroll
  for (int j = 0; j < 8; ++j) {
    int col = j * 16 + row;
#pragma unroll
    for (int v = 0; v < 8; ++v) {
      int m = wm * 16 + v + 8 * half;
      dst[(size_t)(t0 + m) * 128 + col] = acc[j][v];
    }
  }
}

// ---------------------------------------------------------------------------
// K3: causal depthwise conv (kernel 4) + bias + SiLU
// ---------------------------------------------------------------------------
__global__ void k_conv_silu(const float* __restrict__ xs_pre, const float* __restrict__ conv_w,
                            const float* __restrict__ conv_b, float* __restrict__ xs) {
  size_t e = (size_t)blockIdx.x * 256 + threadIdx.x;
  int d = (int)(e & 127);
  size_t t = e >> 7;
  int l = (int)(t & 4095);
  float acc = conv_b[d];
#pragma unroll
  for (int i = 0; i < 4; ++i) {
    int ls = l - 3 + i;
    if (ls >= 0) acc += conv_w[d * 4 + i] * xs_pre[(t - (size_t)(3 - i)) * 128 + d];
  }
  xs[e] = siluf(acc);
}

// ---------------------------------------------------------------------------
// K4: x_dbl[t][k] = sum_d xs[t][d] * Wx[k][d], k<36 (token-major)
// ---------------------------------------------------------------------------
__global__ void k_xdbl(const float* __restrict__ xs, const float* __restrict__ Wx,
                       float* __restrict__ xdbl) {
  size_t e = (size_t)blockIdx.x * 256 + threadIdx.x;
  int k = (int)(e % 36);
  size_t t = e / 36;
  const float* xr = xs + t * 128;
  const float* wr = Wx + (size_t)k * 128;
  float acc = 0.f;
#pragma unroll 8
  for (int d = 0; d < 128; ++d) acc += xr[d] * wr[d];
  xdbl[t * 36 + k] = acc;
}

// ---------------------------------------------------------------------------
// K5: delta[t][d] = softplus(sum_r dt[t][r]*W_dt[d][r] + b_dt[d])
// ---------------------------------------------------------------------------
__global__ void k_delta(const float* __restrict__ xdbl, const float* __restrict__ W_dt,
                        const float* __restrict__ b_dt, float* __restrict__ delta) {
  size_t e = (size_t)blockIdx.x * 256 + threadIdx.x;
  int d = (int)(e & 127);
  size_t t = e >> 7;
  float acc = b_dt[d];
#pragma unroll
  for (int r = 0; r < 4; ++r) acc += xdbl[t * 36 + r] * W_dt[d * 4 + r];
  delta[e] = (acc > 20.f) ? acc : log1pf(__expf(acc));
}

// ---------------------------------------------------------------------------
// K6: selective scan (latency-bound sequential core).
// Wave = (seq s, channels d0=2p, d1=2p+1); lanes 0-15/16-31 = 16 states each.
// ---------------------------------------------------------------------------
__global__ void k_scan(const float* __restrict__ delta, const float* __restrict__ xs,
                       const float* __restrict__ xdbl, const float* __restrict__ A_log,
                       float* __restrict__ ys) {
  int tid  = threadIdx.x;
  int lane = tid & 31;
  int gw   = blockIdx.x * 8 + (tid >> 5);
  int s = gw >> 6;
  int p = gw & 63;
  int n = lane & 15;
  int d = p * 2 + (lane >> 4);
  float A = -__expf(A_log[d * 16 + n]);
  float h = 0.f;
  size_t tbase = (size_t)s * 4096;
  for (int l = 0; l < 4096; ++l) {
    size_t t = tbase + l;
    float dl = delta[t * 128 + d];
    float u  = xs[t * 128 + d];
    float Bn = xdbl[t * 36 + 4 + n];
    float Cn = xdbl[t * 36 + 20 + n];
    h = __expf(dl * A) * h + (dl * u) * Bn;
    float yv = h * Cn;
    yv += __shfl_xor(yv, 1, 16);
    yv += __shfl_xor(yv, 2, 16);
    yv += __shfl_xor(yv, 4, 16);
    yv += __shfl_xor(yv, 8, 16);
    if (n == 0) ys[t * 128 + d] = yv;
  }
}

// ---------------------------------------------------------------------------
// K7: y = (ys + D_skip*xs) * silu(z), in place into ys
// ---------------------------------------------------------------------------
__global__ void k_gate(float* __restrict__ ys, const float* __restrict__ xs,
                       const float* __restrict__ zbuf, const float* __restrict__ D_skip) {
  size_t e = (size_t)blockIdx.x * 256 + threadIdx.x;
  int d = (int)(e & 127);
  float y = ys[e] + D_skip[d] * xs[e];
  ys[e] = y * siluf(zbuf[e]);
}

// ---------------------------------------------------------------------------
// K8: out-projection GEMM. M=NTOK, N=64, K=128 via fp32 WMMA.
// W_out by TDM, A tile by async b128 loads.
// ---------------------------------------------------------------------------
__global__ void k_outproj(const float* __restrict__ y, const float* __restrict__ W_out,
                          float* __restrict__ outb) {
  extern __shared__ float smem[];
  float* sW = smem;              // 64*128 (32 KB)
  float* sA = smem + 64 * 128;   // 64*128 (32 KB)
  int tid = threadIdx.x;
  int t0  = blockIdx.x * 64;

  if (tid < 32) tdm_load_contig(sW, W_out, 64 * 128);
  {
    unsigned lA = (unsigned)(unsigned long long)(void*)sA;
    unsigned g0 = (unsigned)blockIdx.x * (64u * 128u * 4u);
#pragma unroll
    for (unsigned i = 0; i < 8; ++i) {
      unsigned idx = (unsigned)tid + i * 256u;
      async_copy_b128(lA + idx * 16u, g0 + idx * 16u, y);
    }
  }
  wait_async();
  if (tid < 32) __builtin_amdgcn_s_wait_tensorcnt(0);
  __syncthreads();

  int lane = tid & 31, wave = tid >> 5;
  int wm = wave & 3, wn = wave >> 2;
  int row = lane & 15, half = lane >> 4;
  const v8f vz = {0.f, 0.f, 0.f, 0.f, 0.f, 0.f, 0.f, 0.f};
  v8f acc0 = vz, acc1 = vz;
  for (int k0 = 0; k0 < 128; k0 += 4) {
    int kk = k0 + 2 * half;
    v2f a  = *(const v2f*)&sA[(wm * 16 + row) * 128 + kk];
    v2f b0 = *(const v2f*)&sW[(wn * 32 + row) * 128 + kk];
    v2f b1 = *(const v2f*)&sW[(wn * 32 + 16 + row) * 128 + kk];
    acc0 = wmma4(a, b0, acc0);
    acc1 = wmma4(a, b1, acc1);
  }
#pragma unroll
  for (int v = 0; v < 8; ++v) {
    int m = wm * 16 + v + 8 * half;
    outb[(size_t)(t0 + m) * 64 + wn * 32 + row]      = acc0[v];
    outb[(size_t)(t0 + m) * 64 + wn * 32 + 16 + row] = acc1[v];
  }
}

// ---------------------------------------------------------------------------
// K9: direction unscramble + fuse GEMM (K=256, N=64) + residual.
// fuse_w by TDM; gathered A rows (256 B contiguous each) by async b128 loads.
// ---------------------------------------------------------------------------
__global__ void k_fuse(const float* __restrict__ outb, const float* __restrict__ fuse_w,
                       const float* __restrict__ x, const float* __restrict__ scale,
                       float* __restrict__ dst) {
  extern __shared__ float smem[];
  float* sW = smem;              // 64*256 (64 KB)
  float* sA = smem + 64 * 256;   // 64*256 (64 KB)
  int tid = threadIdx.x;
  int blk = blockIdx.x;
  int b = blk >> 6, h = blk & 63;

  if (tid < 32) tdm_load_contig(sW, fuse_w, 64 * 256);
  {
    int dir = tid & 3;
    int w   = tid >> 2;
    int l  = h * 64 + w;
    int lv = w * 64 + h;
    unsigned srow;
    if (dir == 0)      srow = (unsigned)(b * 4096 + l);
    else if (dir == 1) srow = (unsigned)((8 + b) * 4096 + lv);
    else if (dir == 2) srow = (unsigned)((16 + b) * 4096 + (4095 - l));
    else               srow = (unsigned)((24 + b) * 4096 + (4095 - lv));
    unsigned goff = srow * 256u;  // 64 floats per row
    unsigned lA = (unsigned)(unsigned long long)(void*)(sA + (size_t)w * 256 + dir * 64);
#pragma unroll
    for (unsigned i = 0; i < 16; ++i)
      async_copy_b128(lA + i * 16u, goff + i * 16u, outb);
  }
  wait_async();
  if (tid < 32) __builtin_amdgcn_s_wait_tensorcnt(0);
  __syncthreads();

  int lane = tid & 31, wave = tid >> 5;
  int wm = wave & 3, wn = wave >> 2;
  int row = lane & 15, half = lane >> 4;
  const v8f vz = {0.f, 0.f, 0.f, 0.f, 0.f, 0.f, 0.f, 0.f};
  v8f acc0 = vz, acc1 = vz;
  for (int k0 = 0; k0 < 256; k0 += 4) {
    int kk = k0 + 2 * half;
    v2f a  = *(const v2f*)&sA[(wm * 16 + row) * 256 + kk];
    v2f b0 = *(const v2f*)&sW[(wn * 32 + row) * 256 + kk];
    v2f b1 = *(const v2f*)&sW[(wn * 32 + 16 + row) * 256 + kk];
    acc0 = wmma4(a, b0, acc0);
    acc1 = wmma4(a, b1, acc1);
  }
  float sc = scale[0];
#pragma unroll
  for (int v = 0; v < 8; ++v) {
    int w  = wm * 16 + v + 8 * half;
    int c0 = wn * 32 + row;
    int c1 = c0 + 16;
    size_t o0 = ((size_t)(b * 64 + c0) * 64 + h) * 64 + w;
    size_t o1 = ((size_t)(b * 64 + c1) * 64 + h) * 64 + w;
    dst[o0] = x[o0] + sc * acc0[v];
    dst[o1] = x[o1] + sc * acc1[v];
  }
}

// ---------------------------------------------------------------------------
extern "C" void kernel_launch(void* const* d_in, const int* in_sizes, int n_in,
                              void* d_out, int out_size, void* d_ws, size_t ws_size,
                              hipStream_t stream) {
  const float* x      = (const float*)d_in[0];
  const float* ln_w   = (const float*)d_in[1];
  const float* ln_b   = (const float*)d_in[2];
  const float* W_in   = (const float*)d_in[3];
  const float* conv_w = (const float*)d_in[4];
  const float* conv_b = (const float*)d_in[5];
  const float* Wx     = (const float*)d_in[6];
  const float* W_dt   = (const float*)d_in[7];
  const float* b_dt   = (const float*)d_in[8];
  const float* A_log  = (const float*)d_in[9];
  const float* D_skip = (const float*)d_in[10];
  const float* W_out  = (const float*)d_in[11];
  const float* fuse_w = (const float*)d_in[12];
  const float* scale  = (const float*)d_in[13];

  float* ws = (float*)d_ws;
  size_t off = 0;
  float* normed = ws + off; off += (size_t)NTOK * 64;    // reused as out-proj result
  float* xs_pre = ws + off; off += (size_t)NTOK * 128;   // reused as ys
  float* zbuf   = ws + off; off += (size_t)NTOK * 128;
  float* xs     = ws + off; off += (size_t)NTOK * 128;
  float* xdbl   = ws + off; off += (size_t)NTOK * 36;
  float* delta  = ws + off; off += (size_t)NTOK * 128;
  float* ys     = xs_pre;   // xs_pre dead after conv
  float* outb   = normed;   // normed dead after in-proj

  k_prep_ln <<<NTOK / 8,            256, 0, stream>>>(x, ln_w, ln_b, normed);
  k_inproj  <<<NTOK / 64,           256, (256 * 64 + 64 * 64) * sizeof(float), stream>>>
            (normed, W_in, xs_pre, zbuf);
  k_conv_silu<<<NTOK * 128 / 256,   256, 0, stream>>>(xs_pre, conv_w, conv_b, xs);
  k_xdbl    <<<NTOK * 36 / 256,     256, 0, stream>>>(xs, Wx, xdbl);
  k_delta   <<<NTOK * 128 / 256,    256, 0, stream>>>(xdbl, W_dt, b_dt, delta);
  k_scan    <<<256,                 256, 0, stream>>>(delta, xs, xdbl, A_log, ys);
  k_gate    <<<NTOK * 128 / 256,    256, 0, stream>>>(ys, xs, zbuf, D_skip);
  k_outproj <<<NTOK / 64,           256, (2 * 64 * 128) * sizeof(float), stream>>>
            (ys, W_out, outb);
  k_fuse    <<<512,                 256, (2 * 64 * 256) * sizeof(float), stream>>>
            (outb, fuse_w, x, scale, (float*)d_out);
}